// HeteroGNN_62088047231456
// MI455X (gfx1250) — compile-verified
//
#include <hip/hip_runtime.h>
#include <hip/hip_bf16.h>

// ---------------------------------------------------------------------------
// HeteroGNN (2x hetero-GAT) forward for MI455X (gfx1250, wave32).
//  * All dense projections use exact-precision V_WMMA_F32_16X16X4_F32.
//  * Edge-feature projection he = ea@W_edge is never materialized: it only
//    appears via a linear functional (he * att_edge).sum, so we pre-fold
//    W_edge/att_edge into w_ae[64,2] per layer and fuse with the edge MLP.
//  * Edge-MLP weights staged into LDS with GLOBAL_LOAD_ASYNC_TO_LDS_B128
//    (ASYNCcnt path) instead of a VGPR round-trip.
//  * Segment softmax via f32 atomics; accumulators fit in the 192MB L2.
// ---------------------------------------------------------------------------

typedef float v2f __attribute__((ext_vector_type(2)));
typedef float v8f __attribute__((ext_vector_type(8)));

// ---------------- generic fill ----------------
__global__ void fillk(float* __restrict__ p, float v, long long n) {
  long long i = (long long)blockIdx.x * blockDim.x + threadIdx.x;
  if (i < n) p[i] = v;
}

// ---------------- WMMA f32 GEMM:  Y[M,N] = act(X[M,K] @ W[K,N] + bias) ----
// block = 128 threads = 4 waves; each wave computes one 16x16 tile.
// grid = (M/16, N/64). M % 16 == 0, N % 64 == 0, K % 4 == 0 (guaranteed).
template <int ACT>   // 0 = none, 1 = leaky-relu 0.01
__global__ void gemm_wmma(const float* __restrict__ X, const float* __restrict__ W,
                          const float* __restrict__ bias, float* __restrict__ Y,
                          int M, int K, int N) {
  const int wave = threadIdx.x >> 5;
  const int lane = threadIdx.x & 31;
  const int m0 = blockIdx.x * 16;
  const int n0 = blockIdx.y * 64 + wave * 16;
  const int row = m0 + (lane & 15);         // A: lanes 0-15 and 16-31 both M=0..15
  const int khalf = (lane >> 4) * 2;        // A/B: K 0,1 in low lanes; 2,3 in high
  const int col = n0 + (lane & 15);         // B/C/D: N striped across 16 lanes
  v8f acc = {};
  const float* xr = X + (long long)row * K + khalf;
  const float* wp = W + (long long)khalf * N + col;
  const long long wstep = 4LL * N;          // uniform stride per K-step
  for (int k = 0; k < K; k += 4) {
    v2f a, b;
    a[0] = xr[k];                           // X[row][k+khalf]
    a[1] = xr[k + 1];                       // X[row][k+khalf+1]
    b[0] = wp[0];                           // W[k+khalf][col]
    b[1] = wp[N];                           // W[k+khalf+1][col]
    acc = __builtin_amdgcn_wmma_f32_16x16x4_f32(false, a, false, b,
                                                (short)0, acc, false, false);
    wp += wstep;
  }
  const float bv = bias ? bias[col] : 0.0f;
  const int rbase = m0 + ((lane >> 4) << 3);  // C: VGPR r -> M=r (lo) / r+8 (hi)
#pragma unroll
  for (int r = 0; r < 8; ++r) {
    float v = acc[r] + bv;
    if (ACT) v = v > 0.0f ? v : 0.01f * v;
    Y[(long long)(rbase + r) * N + col] = v;
  }
}

// ---------------- fold W_edge & att_edge into w_ae[64,4] (2 layers) -------
__global__ void make_wae(const float* __restrict__ W_edge, const float* __restrict__ att_edge,
                         float* __restrict__ wae, int HC, int C, int slot) {
  int t = blockIdx.x * blockDim.x + threadIdx.x;   // 128 threads: (k, h)
  if (t >= 128) return;
  int k = t >> 1, h = t & 1;
  float s = 0.0f;
  for (int c = 0; c < C; ++c)
    s += W_edge[(long long)k * HC + h * C + c] * att_edge[h * C + c];
  wae[k * 4 + slot + h] = s;
}

// ---------------- fused edge MLP + attention-edge dot ----------------------
// ae[e, 0..3] = lrelu01(ea0[e] @ We + be) . w_ae[:, 0..3]   (layers 1&2, heads 0&1)
// optionally accumulates segment sums (for self-loop mean) and counts.
// We[16,64] (4KB) is staged into LDS via the CDNA5 async global->LDS path:
// 256 threads x one GLOBAL_LOAD_ASYNC_TO_LDS_B128 each, tracked by ASYNCcnt.
__global__ void edge_ae(const float* __restrict__ ea0, const float* __restrict__ We,
                        const float* __restrict__ be, const float* __restrict__ wae,
                        const int* __restrict__ dst,
                        float* __restrict__ ae, float* __restrict__ sum_ae,
                        float* __restrict__ cnt, long long E) {
  __shared__ __align__(16) float sW[16 * 64];
  __shared__ float sb[64];
  __shared__ float swae[64 * 4];
  {
    // async copy: 256 threads * 16B = 4KB = whole We block, no VGPR round-trip.
    // LDS aperture: low 32 bits of the generic shared pointer == LDS byte offset.
    unsigned lds = (unsigned)(unsigned long long)(uintptr_t)(&sW[threadIdx.x * 4]);
    unsigned long long ga = (unsigned long long)(uintptr_t)(We + threadIdx.x * 4);
    asm volatile("global_load_async_to_lds_b128 %0, %1, off"
                 :: "v"(lds), "v"(ga) : "memory");
  }
  if (threadIdx.x < 64) sb[threadIdx.x] = be[threadIdx.x];
  for (int i = threadIdx.x; i < 256; i += blockDim.x) swae[i] = wae[i];
  asm volatile("s_wait_asynccnt 0" ::: "memory");
  __syncthreads();
  long long e = (long long)blockIdx.x * blockDim.x + threadIdx.x;
  if (e >= E) return;
  float x[16];
#pragma unroll
  for (int k = 0; k < 16; ++k) x[k] = ea0[e * 16 + k];
  float r0 = 0, r1 = 0, r2 = 0, r3 = 0;
  for (int j = 0; j < 64; ++j) {
    float t = sb[j];
#pragma unroll
    for (int k = 0; k < 16; ++k) t += x[k] * sW[k * 64 + j];
    t = t > 0.0f ? t : 0.01f * t;
    r0 += t * swae[j * 4 + 0];
    r1 += t * swae[j * 4 + 1];
    r2 += t * swae[j * 4 + 2];
    r3 += t * swae[j * 4 + 3];
  }
  ae[e * 4 + 0] = r0; ae[e * 4 + 1] = r1; ae[e * 4 + 2] = r2; ae[e * 4 + 3] = r3;
  if (sum_ae) {
    int d = dst[e];
    atomicAdd(&sum_ae[(long long)d * 4 + 0], r0);
    atomicAdd(&sum_ae[(long long)d * 4 + 1], r1);
    atomicAdd(&sum_ae[(long long)d * 4 + 2], r2);
    atomicAdd(&sum_ae[(long long)d * 4 + 3], r3);
    atomicAdd(&cnt[d], 1.0f);
  }
}

// ae_loop[n,j] = sum_ae[n,j] / max(cnt[n],1)   (in place)
__global__ void div_mean(float* __restrict__ sum_ae, const float* __restrict__ cnt, long long n4) {
  long long i = (long long)blockIdx.x * blockDim.x + threadIdx.x;
  if (i >= n4) return;
  sum_ae[i] = sum_ae[i] / fmaxf(cnt[i >> 2], 1.0f);
}

// out[n, j] = b1[j] (+ b2[j])
__global__ void bias_init(float* __restrict__ out, const float* __restrict__ b1,
                          const float* __restrict__ b2, long long total, int HC) {
  long long i = (long long)blockIdx.x * blockDim.x + threadIdx.x;
  if (i >= total) return;
  int j = (int)(i % HC);
  out[i] = b1[j] + (b2 ? b2[j] : 0.0f);
}

// out[n,h] = sum_c h[n, h*C+c] * att[h*C+c]
__global__ void node_att_dot(const float* __restrict__ h, const float* __restrict__ att,
                             float* __restrict__ out, long long n, int H, int C) {
  long long i = (long long)blockIdx.x * blockDim.x + threadIdx.x;
  if (i >= n * H) return;
  long long node = i / H; int hh = (int)(i % H);
  const float* hp = h + node * H * C + (long long)hh * C;
  const float* ap = att + hh * C;
  float s = 0.0f;
  for (int c = 0; c < C; ++c) s += hp[c] * ap[c];
  out[i] = s;
}

__device__ inline void atomicMaxF(float* addr, float val) {
  int* ia = (int*)addr;
  int old = __float_as_int(*addr);
  while (__int_as_float(old) < val) {
    int assumed = old;
    old = atomicCAS(ia, assumed, __float_as_int(val));
    if (old == assumed) break;
  }
}

// self-loop raw score; also initializes amax (= loop score, every dst has one)
__global__ void loop_score(const float* __restrict__ as_, const float* __restrict__ ad_,
                           const float* __restrict__ ae_loop, int layer,
                           float* __restrict__ aloop, float* __restrict__ amax, long long n) {
  long long i = (long long)blockIdx.x * blockDim.x + threadIdx.x;
  if (i >= n * 2) return;
  long long node = i >> 1; int h = (int)(i & 1);
  float a = as_[i] + ad_[i] + ae_loop[node * 4 + layer * 2 + h];
  a = a > 0.0f ? a : 0.2f * a;
  aloop[i] = a;
  amax[i] = a;
}

__global__ void edge_score(const float* __restrict__ as_, const float* __restrict__ ad_,
                           const float* __restrict__ ae, int layer,
                           const int* __restrict__ src, const int* __restrict__ dst,
                           float* __restrict__ a, float* __restrict__ amax, long long E) {
  long long i = (long long)blockIdx.x * blockDim.x + threadIdx.x;
  if (i >= E * 2) return;
  long long e = i >> 1; int h = (int)(i & 1);
  int s = src[e], d = dst[e];
  float v = as_[(long long)s * 2 + h] + ad_[(long long)d * 2 + h] + ae[e * 4 + layer * 2 + h];
  v = v > 0.0f ? v : 0.2f * v;
  a[i] = v;
  atomicMaxF(&amax[(long long)d * 2 + h], v);
}

// aloop = exp(aloop - amax); den = aloop   (loop-edge seeds the denominator)
__global__ void loop_exp(float* __restrict__ aloop, const float* __restrict__ amax,
                         float* __restrict__ den, long long n2) {
  long long i = (long long)blockIdx.x * blockDim.x + threadIdx.x;
  if (i >= n2) return;
  float ex = expf(aloop[i] - amax[i]);
  aloop[i] = ex;
  den[i] = ex;
}

__global__ void edge_exp(float* __restrict__ a, const float* __restrict__ amax,
                         const int* __restrict__ dst, float* __restrict__ den, long long E) {
  long long i = (long long)blockIdx.x * blockDim.x + threadIdx.x;
  if (i >= E * 2) return;
  long long e = i >> 1; int h = (int)(i & 1);
  int d = dst[e];
  float ex = expf(a[i] - amax[(long long)d * 2 + h]);
  a[i] = ex;
  atomicAdd(&den[(long long)d * 2 + h], ex);
}

// out[n, j] += (aloop/den) * h[n, j]   (self-loop aggregation, src == dst)
__global__ void loop_aggr(float* __restrict__ out, const float* __restrict__ h,
                          const float* __restrict__ aloop, const float* __restrict__ den,
                          long long n, int H, int C) {
  long long total = n * H * C;
  long long i = (long long)blockIdx.x * blockDim.x + threadIdx.x;
  if (i >= total) return;
  long long node = i / (H * C); int j = (int)(i % (H * C)); int hh = j / C;
  float al = aloop[node * H + hh] / fmaxf(den[node * H + hh], 1e-16f);
  out[i] += al * h[i];
}

// out[dst, j] += alpha * hs[src, j]
__global__ void edge_aggr(float* __restrict__ out, const float* __restrict__ hs,
                          const float* __restrict__ ex, const float* __restrict__ den,
                          const int* __restrict__ src, const int* __restrict__ dst,
                          long long E, int H, int C) {
  long long total = E * H * C;
  long long i = (long long)blockIdx.x * blockDim.x + threadIdx.x;
  if (i >= total) return;
  long long e = i / (H * C); int j = (int)(i % (H * C)); int hh = j / C;
  int s = src[e], d = dst[e];
  float al = ex[e * H + hh] / fmaxf(den[(long long)d * H + hh], 1e-16f);
  atomicAdd(&out[(long long)d * H * C + j], al * hs[(long long)s * H * C + j]);
}

// final: out[n] = xp3[n,:] . W[:,0] + b
__global__ void final_out(const float* __restrict__ x, const float* __restrict__ W,
                          const float* __restrict__ b, float* __restrict__ out,
                          long long n, int K) {
  long long i = (long long)blockIdx.x * blockDim.x + threadIdx.x;
  if (i >= n) return;
  float s = b[0];
  const float* xp = x + i * K;
  for (int k = 0; k < K; ++k) s += xp[k] * W[k];
  out[i] = s;
}

// ---------------------------------------------------------------------------
extern "C" void kernel_launch(void* const* d_in, const int* in_sizes, int n_in,
                              void* d_out, int out_size, void* d_ws, size_t ws_size,
                              hipStream_t stream) {
  if (n_in < 59) return;
  auto F = [&](int i) { return (const float*)d_in[i]; };
  auto I = [&](int i) { return (const int*)d_in[i]; };

  // params flattened: jax pytree order (sorted keys at every dict level).
  // conv1.bb {W, W_edge, att_dst, att_edge, att_src, bias} -> 0..5
  // conv1.bp {W_dst, W_edge, W_src, att_dst, att_edge, att_src, bias} -> 6..12
  // conv1.pp -> 13..18 ; conv2.bb -> 19..24 ; conv2.bp -> 25..31 ; conv2.pp -> 32..37
  // e_bb {W,b} 38,39 ; e_bp 40,41 ; e_pp 42,43 ; in_b 44,45 ; in_p 46,47 ; out 48,49
  const long long NP = in_sizes[50] / 32;
  const long long NB = in_sizes[51] / 32;
  const long long EPP = in_sizes[52] / 2;
  const long long EBB = in_sizes[53] / 2;
  const long long EBP = in_sizes[54];
  const float* x_p = F(50);
  const float* x_b = F(51);
  const int* pp_src = I(52);            const int* pp_dst = I(52) + EPP;
  const int* bb_src = I(53);            const int* bb_dst = I(53) + EBB;
  const int* bp_src = I(54);            const int* bp_dst = I(55);
  const float* ea_pp = F(56);
  const float* ea_bb = F(57);
  const float* ea_bp = F(58);

  // -------- workspace arena (floats) --------
  float* w = (float*)d_ws;
  auto alloc = [&](long long n) { float* p = w; w += n; return p; };
  float* xp1   = alloc(NP * 64);    // lrelu(x_p @ in_p)
  float* xb1   = alloc(NB * 64);
  float* xp2   = alloc(NP * 128);   // conv1 out_p
  float* xb2   = alloc(NB * 128);   // conv1 out_b
  float* xp3   = alloc(NP * 256);   // conv2 out_p
  float* hsbuf = alloc(NB * 256);   // projected src feats (reused per relation)
  float* hdbuf = alloc(NP * 256);   // projected dst feats (bp only)
  float* aePP  = alloc(EPP * 4);    // [l1h0,l1h1,l2h0,l2h1] per edge
  float* aeBB  = alloc(EBB * 4);
  float* aeBP  = alloc(EBP * 4);
  float* aelPP = alloc(NP * 4);     // self-loop mean ae (both layers)
  float* aelBB = alloc(NB * 4);
  float* cntPP = alloc(NP);
  float* cntBB = alloc(NB);
  float* asb   = alloc(NB * 2);
  float* adb   = alloc(NB * 2);
  float* amaxb = alloc(NB * 2);
  float* denb  = alloc(NB * 2);
  float* alpb  = alloc(NB * 2);
  float* aedge = alloc(EBB * 2);    // per-edge score/exp scratch (max E)
  float* waePP = alloc(256);
  float* waeBB = alloc(256);
  float* waeBP = alloc(256);
  (void)ws_size;

  auto nb = [](long long n, int t) { return dim3((unsigned)((n + t - 1) / t)); };
  auto gemm = [&](const float* X, const float* W_, const float* b, float* Y,
                  int M, int K, int N, bool act) {
    dim3 g(M / 16, N / 64);
    if (act) gemm_wmma<1><<<g, 128, 0, stream>>>(X, W_, b, Y, M, K, N);
    else     gemm_wmma<0><<<g, 128, 0, stream>>>(X, W_, b, Y, M, K, N);
  };

  // -------- stage 0: input MLPs + fused edge-attention precompute --------
  gemm(x_p, F(46), F(47), xp1, (int)NP, 32, 64, true);
  gemm(x_b, F(44), F(45), xb1, (int)NB, 32, 64, true);

  // fold W_edge/att_edge: conv1 HC=128 C=64, conv2 HC=256 C=128
  make_wae<<<1, 128, 0, stream>>>(F(14), F(16), waePP, 128, 64, 0);
  make_wae<<<1, 128, 0, stream>>>(F(33), F(35), waePP, 256, 128, 2);
  make_wae<<<1, 128, 0, stream>>>(F(1),  F(3),  waeBB, 128, 64, 0);
  make_wae<<<1, 128, 0, stream>>>(F(20), F(22), waeBB, 256, 128, 2);
  make_wae<<<1, 128, 0, stream>>>(F(7),  F(10), waeBP, 128, 64, 0);
  make_wae<<<1, 128, 0, stream>>>(F(26), F(28), waeBP, 256, 128, 2);

  fillk<<<nb(NP * 4, 256), 256, 0, stream>>>(aelPP, 0.0f, NP * 4);
  fillk<<<nb(NP, 256), 256, 0, stream>>>(cntPP, 0.0f, NP);
  fillk<<<nb(NB * 4, 256), 256, 0, stream>>>(aelBB, 0.0f, NB * 4);
  fillk<<<nb(NB, 256), 256, 0, stream>>>(cntBB, 0.0f, NB);

  edge_ae<<<nb(EPP, 256), 256, 0, stream>>>(ea_pp, F(42), F(43), waePP, pp_dst, aePP, aelPP, cntPP, EPP);
  edge_ae<<<nb(EBB, 256), 256, 0, stream>>>(ea_bb, F(38), F(39), waeBB, bb_dst, aeBB, aelBB, cntBB, EBB);
  edge_ae<<<nb(EBP, 256), 256, 0, stream>>>(ea_bp, F(40), F(41), waeBP, nullptr, aeBP, nullptr, nullptr, EBP);
  div_mean<<<nb(NP * 4, 256), 256, 0, stream>>>(aelPP, cntPP, NP * 4);
  div_mean<<<nb(NB * 4, 256), 256, 0, stream>>>(aelBB, cntBB, NB * 4);

  // -------- generic GAT pipeline --------
  auto run_gat = [&](const float* hs, const float* hd, const float* att_src,
                     const float* att_dst, const float* ae, const float* aeloop,
                     int layer, const int* src, const int* dst, long long E,
                     float* out, long long n_src, long long n_dst, int C) {
    node_att_dot<<<nb(n_src * 2, 256), 256, 0, stream>>>(hs, att_src, asb, n_src, 2, C);
    node_att_dot<<<nb(n_dst * 2, 256), 256, 0, stream>>>(hd, att_dst, adb, n_dst, 2, C);
    if (aeloop) {
      loop_score<<<nb(n_dst * 2, 256), 256, 0, stream>>>(asb, adb, aeloop, layer, alpb, amaxb, n_dst);
    } else {
      fillk<<<nb(n_dst * 2, 256), 256, 0, stream>>>(amaxb, -3.0e38f, n_dst * 2);
      fillk<<<nb(n_dst * 2, 256), 256, 0, stream>>>(denb, 0.0f, n_dst * 2);
    }
    edge_score<<<nb(E * 2, 256), 256, 0, stream>>>(asb, adb, ae, layer, src, dst, aedge, amaxb, E);
    if (aeloop)
      loop_exp<<<nb(n_dst * 2, 256), 256, 0, stream>>>(alpb, amaxb, denb, n_dst * 2);
    edge_exp<<<nb(E * 2, 256), 256, 0, stream>>>(aedge, amaxb, dst, denb, E);
    if (aeloop)
      loop_aggr<<<nb(n_dst * 2 * C, 256), 256, 0, stream>>>(out, hs, alpb, denb, n_dst, 2, C);
    edge_aggr<<<nb(E * 2 * C, 256), 256, 0, stream>>>(out, hs, aedge, denb, src, dst, E, 2, C);
  };

  // -------- conv1 (H=2, C=64, HC=128) --------
  bias_init<<<nb(NP * 128, 256), 256, 0, stream>>>(xp2, F(18), F(12), NP * 128, 128); // pp.bias + bp.bias
  bias_init<<<nb(NB * 128, 256), 256, 0, stream>>>(xb2, F(5), nullptr, NB * 128, 128);

  gemm(xp1, F(13), nullptr, hsbuf, (int)NP, 64, 128, false);                    // pp: h = xp1 @ W
  run_gat(hsbuf, hsbuf, F(17), F(15), aePP, aelPP, 0, pp_src, pp_dst, EPP, xp2, NP, NP, 64);

  gemm(xb1, F(0), nullptr, hsbuf, (int)NB, 64, 128, false);                     // bb
  run_gat(hsbuf, hsbuf, F(4), F(2), aeBB, aelBB, 0, bb_src, bb_dst, EBB, xb2, NB, NB, 64);

  gemm(xb1, F(8), nullptr, hsbuf, (int)NB, 64, 128, false);                     // bp: hs = xb1 @ W_src
  gemm(xp1, F(6), nullptr, hdbuf, (int)NP, 64, 128, false);                     //     hd = xp1 @ W_dst
  run_gat(hsbuf, hdbuf, F(11), F(9), aeBP, nullptr, 0, bp_src, bp_dst, EBP, xp2, NB, NP, 64);

  // -------- conv2 (H=2, C=128, HC=256; bb output is dead -> skipped) --------
  bias_init<<<nb(NP * 256, 256), 256, 0, stream>>>(xp3, F(37), F(31), NP * 256, 256);

  gemm(xp2, F(32), nullptr, hsbuf, (int)NP, 128, 256, false);                   // pp
  run_gat(hsbuf, hsbuf, F(36), F(34), aePP, aelPP, 1, pp_src, pp_dst, EPP, xp3, NP, NP, 128);

  gemm(xb2, F(27), nullptr, hsbuf, (int)NB, 128, 256, false);                   // bp hs
  gemm(xp2, F(25), nullptr, hdbuf, (int)NP, 128, 256, false);                   // bp hd
  run_gat(hsbuf, hdbuf, F(30), F(29), aeBP, nullptr, 1, bp_src, bp_dst, EBP, xp3, NB, NP, 128);

  // -------- final linear [256 -> 1] --------
  final_out<<<nb(NP, 256), 256, 0, stream>>>(xp3, F(48), F(49), (float*)d_out, NP, 256);
}